// PPNet_86157043958180
// MI455X (gfx1250) — compile-verified
//
#include <hip/hip_runtime.h>
#include <math.h>

// CDNA5 / gfx1250 WMMA fragment types for V_WMMA_F32_16X16X4_F32 (wave32):
//  A (16x4 f32)  = 64 vals / 32 lanes = 2 VGPRs -> float2
//  B (4x16 f32)  = 2 VGPRs -> float2
//  C/D (16x16)   = 8 VGPRs -> float8
typedef __attribute__((ext_vector_type(2))) float v2f;
typedef __attribute__((ext_vector_type(8))) float v8f;

#define EPSV 1e-4f
#define BIGF 3.4e38f

// ---------------------------------------------------------------------------
// Stage 1: conv1 (Cin=1, k=7, s=2)  x:(1280,1,2048) -> h1:(1280,64,1021), ReLU
// K=7 not a multiple of 4 and ~0.7% of total FLOPs -> plain VALU kernel.
// ---------------------------------------------------------------------------
__global__ __launch_bounds__(256)
void conv1_kernel(const float* __restrict__ x, const float* __restrict__ w1,
                  const float* __restrict__ b1, float* __restrict__ out)
{
    const int L_IN = 2048, L_OUT = 1021;
    int l  = blockIdx.x * blockDim.x + threadIdx.x;
    int co = blockIdx.y;
    int n  = blockIdx.z;
    if (l >= L_OUT) return;
    const float* xn = x + (size_t)n * L_IN;
    float s = b1[co];
    #pragma unroll
    for (int k = 0; k < 7; k++) s += xn[2 * l + k] * w1[co * 7 + k];
    out[(size_t)n * (64 * 1021) + (size_t)co * 1021 + l] = fmaxf(s, 0.0f);
}

// ---------------------------------------------------------------------------
// Conv-as-GEMM via WMMA f32 16x16x4, register-blocked over ALL output-channel
// tiles: one wave owns a full 16-column slice (all COUT rows). Each B fragment
// (activations = the expensive operand) is loaded once per k-chunk and reused
// by MT = COUT/16 WMMAs; A fragments come from the tiny cached weight tensors.
// ACT: 0 = ReLU, 1 = sigmoid (+ fused x2sum = sum_c out^2 per column).
// ---------------------------------------------------------------------------
template<int CIN, int KW, int STRIDE, int LIN, int LOUT, int COUT, int ACT>
__global__ __launch_bounds__(32)
void conv_gemm_wmma(const float* __restrict__ in, const float* __restrict__ W,
                    const float* __restrict__ bias, float* __restrict__ out,
                    float* __restrict__ x2 /* used when ACT==1 */)
{
    constexpr int K  = CIN * KW;      // 320 / 384 / 256 / 128 (all %4==0)
    constexpr int MT = COUT / 16;     // 8 / 16 / 8 / 8 tiles, all in registers
    const int n    = blockIdx.x;
    const int lT   = blockIdx.y * 16;
    const int lane = threadIdx.x;
    const int half = lane >> 4;
    const int lm   = lane & 15;

    const float* inN = in + (size_t)n * CIN * LIN;
    const int l  = lT + lm;
    const int lc = (l < LOUT) ? l : (LOUT - 1);   // clamp; stores masked later

    const v8f vzero = {};
    v8f acc[MT];
    #pragma unroll
    for (int mt = 0; mt < MT; mt++) acc[mt] = vzero;

    #pragma unroll 1
    for (int kk = 0; kk < K; kk += 4) {
        const int k0 = kk + half * 2;
        // B fragment: im2col column lc, k-rows k0,k0+1 (this lane's slots)
        const int ci0 = k0 / KW,        kw0 = k0 - ci0 * KW;
        const int ci1 = (k0 + 1) / KW,  kw1 = (k0 + 1) - ci1 * KW;
        v2f b;
        b[0] = inN[(size_t)ci0 * LIN + STRIDE * lc + kw0];
        b[1] = inN[(size_t)ci1 * LIN + STRIDE * lc + kw1];
        const float* Wk = W + (size_t)lm * K + k0;
        #pragma unroll
        for (int mt = 0; mt < MT; mt++) {
            v2f a;
            a[0] = Wk[(size_t)mt * 16 * K];
            a[1] = Wk[(size_t)mt * 16 * K + 1];
            acc[mt] = __builtin_amdgcn_wmma_f32_16x16x4_f32(
                false, a, false, b, (short)0, acc[mt], false, false);
        }
    }

    float* outN = out + (size_t)n * COUT * LOUT;
    float ssq = 0.0f;
    if (l < LOUT) {
        #pragma unroll
        for (int mt = 0; mt < MT; mt++) {
            #pragma unroll
            for (int r = 0; r < 8; r++) {
                const int co = mt * 16 + r + half * 8;
                float v = acc[mt][r] + bias[co];
                if (ACT == 0) v = fmaxf(v, 0.0f);
                else          v = 1.0f / (1.0f + expf(-v));
                outN[(size_t)co * LOUT + l] = v;
                if (ACT == 1) ssq += v * v;
            }
        }
    }
    if (ACT == 1) {
        // half 0 holds rows mt*16+0..7, half 1 rows mt*16+8..15: combine them.
        ssq += __shfl_xor(ssq, 16, 32);
        if (half == 0 && l < LOUT) x2[(size_t)n * LOUT + l] = ssq;
    }
}

// p2[p] = sum_d protos[p][d]^2
__global__ __launch_bounds__(256)
void p2_kernel(const float* __restrict__ protos, float* __restrict__ p2)
{
    const int P = 200, D = 128;
    int p = threadIdx.x;
    if (p >= P) return;
    float s = 0.0f;
    for (int d = 0; d < D; d++) { float v = protos[p * D + d]; s += v * v; }
    p2[p] = s;
}

// ---------------------------------------------------------------------------
// Prototype distances + min over L, fused, register-blocked over MT p-tiles
// (P=200 split across two waves: tiles [0,7) and [7,13) so VGPRs stay sane).
// xp = protos(200x128) x h[n](128x254) via WMMA; dist = relu(x2 - 2*xp + p2);
// running min over columns, then cross-lane min inside each 16-lane half.
// min_out[n*200+p]  (== min_d reshaped (64, 20*200) flat)
// ---------------------------------------------------------------------------
template<int MT, int PBASE>
__global__ __launch_bounds__(32)
void proto_min_wmma(const float* __restrict__ h, const float* __restrict__ protos,
                    const float* __restrict__ x2sum, const float* __restrict__ p2,
                    float* __restrict__ min_out)
{
    const int D = 128, L = 254, P = 200;
    const int n    = blockIdx.x;
    const int lane = threadIdx.x;
    const int half = lane >> 4;
    const int lm   = lane & 15;
    const float* hn = h + (size_t)n * D * L;

    float mn[MT * 8];
    #pragma unroll
    for (int i = 0; i < MT * 8; i++) mn[i] = BIGF;

    const v8f vzero = {};

    #pragma unroll 1
    for (int lT = 0; lT < L; lT += 16) {
        const int l  = lT + lm;
        const int lc = (l < L) ? l : (L - 1);

        v8f acc[MT];
        #pragma unroll
        for (int mt = 0; mt < MT; mt++) acc[mt] = vzero;

        #pragma unroll 1
        for (int kk = 0; kk < D; kk += 4) {
            const int k0 = kk + half * 2;
            v2f b;
            b[0] = hn[(size_t)k0 * L + lc];
            b[1] = hn[(size_t)(k0 + 1) * L + lc];
            #pragma unroll
            for (int mt = 0; mt < MT; mt++) {
                const int rowRaw = PBASE * 16 + mt * 16 + lm;
                const int row    = (rowRaw < P) ? rowRaw : (P - 1);
                v2f a;
                a[0] = protos[row * D + k0];
                a[1] = protos[row * D + k0 + 1];
                acc[mt] = __builtin_amdgcn_wmma_f32_16x16x4_f32(
                    false, a, false, b, (short)0, acc[mt], false, false);
            }
        }

        const float xs    = x2sum[(size_t)n * L + lc];
        const bool  valid = (l < L);
        #pragma unroll
        for (int mt = 0; mt < MT; mt++) {
            #pragma unroll
            for (int r = 0; r < 8; r++) {
                const int p  = PBASE * 16 + mt * 16 + r + half * 8;
                const int pc = (p < P) ? p : (P - 1);
                float d = xs - 2.0f * acc[mt][r] + p2[pc];
                d = fmaxf(d, 0.0f);
                if (!valid) d = BIGF;
                mn[mt * 8 + r] = fminf(mn[mt * 8 + r], d);
            }
        }
    }

    // min across the 16 lanes of each half (masks 8,4,2,1 stay inside a half)
    #pragma unroll
    for (int i = 0; i < MT * 8; i++) {
        #pragma unroll
        for (int m = 8; m >= 1; m >>= 1)
            mn[i] = fminf(mn[i], __shfl_xor(mn[i], m, 32));
    }

    if (lm == 0) {
        #pragma unroll
        for (int mt = 0; mt < MT; mt++) {
            #pragma unroll
            for (int r = 0; r < 8; r++) {
                const int p = PBASE * 16 + mt * 16 + r + half * 8;
                if (p < P) min_out[(size_t)n * P + p] = mn[mt * 8 + r];
            }
        }
    }
}

// ---------------------------------------------------------------------------
// outputs[b][c] = sum_j log((md+1)/(md+EPS)) * last_w[c][j], j in [0,4000)
// ---------------------------------------------------------------------------
__global__ __launch_bounds__(256)
void final_kernel(const float* __restrict__ min_d, const float* __restrict__ last_w,
                  float* __restrict__ out)
{
    const int J = 4000;
    int b = blockIdx.x, t = threadIdx.x;
    float s0 = 0.0f, s1 = 0.0f;
    for (int j = t; j < J; j += 256) {
        float md = min_d[(size_t)b * J + j];
        float a  = logf((md + 1.0f) / (md + EPSV));
        s0 += a * last_w[j];
        s1 += a * last_w[J + j];
    }
    __shared__ float r0[256], r1[256];
    r0[t] = s0; r1[t] = s1;
    __syncthreads();
    for (int ofs = 128; ofs > 0; ofs >>= 1) {
        if (t < ofs) { r0[t] += r0[t + ofs]; r1[t] += r1[t + ofs]; }
        __syncthreads();
    }
    if (t == 0) { out[b * 2 + 0] = r0[0]; out[b * 2 + 1] = r1[0]; }
}

// ---------------------------------------------------------------------------
// Launch: ping-pong activations through d_ws.
//   bufA: max(h1 64*1021, h3 256*254, h5 128*254) per sample = 65344 floats
//   bufB: max(h2 128*509, h4 128*254)             per sample = 65152 floats
// Total ws ~670 MB.
// ---------------------------------------------------------------------------
extern "C" void kernel_launch(void* const* d_in, const int* in_sizes, int n_in,
                              void* d_out, int out_size, void* d_ws, size_t ws_size,
                              hipStream_t stream)
{
    const float* x      = (const float*)d_in[0];
    // d_in[1] = metadata (unused, matches reference)
    const float* w1     = (const float*)d_in[2];
    const float* b1     = (const float*)d_in[3];
    const float* w2     = (const float*)d_in[4];
    const float* b2     = (const float*)d_in[5];
    const float* w3     = (const float*)d_in[6];
    const float* b3     = (const float*)d_in[7];
    const float* a1w    = (const float*)d_in[8];
    const float* a1b    = (const float*)d_in[9];
    const float* a2w    = (const float*)d_in[10];
    const float* a2b    = (const float*)d_in[11];
    const float* protos = (const float*)d_in[12];
    const float* lastw  = (const float*)d_in[13];
    float* out = (float*)d_out;                    // [0,128): outputs; [128,...): min_d

    float* bufA = (float*)d_ws;
    float* bufB = bufA + (size_t)1280 * 64 * 1021;
    float* x2   = bufB + (size_t)1280 * 128 * 509;
    float* p2b  = x2   + (size_t)1280 * 254;

    // conv1 -> bufA (1280,64,1021)
    conv1_kernel<<<dim3(4, 64, 1280), 256, 0, stream>>>(x, w1, b1, bufA);
    // conv2 -> bufB (1280,128,509); 32 column tiles
    conv_gemm_wmma<64, 5, 2, 1021, 509, 128, 0>
        <<<dim3(1280, 32), 32, 0, stream>>>(bufA, w2, b2, bufB, nullptr);
    // conv3 -> bufA (1280,256,254); 16 column tiles
    conv_gemm_wmma<128, 3, 2, 509, 254, 256, 0>
        <<<dim3(1280, 16), 32, 0, stream>>>(bufB, w3, b3, bufA, nullptr);
    // a1 (1x1) -> bufB (1280,128,254)
    conv_gemm_wmma<256, 1, 1, 254, 254, 128, 0>
        <<<dim3(1280, 16), 32, 0, stream>>>(bufA, a1w, a1b, bufB, nullptr);
    // a2 (1x1, sigmoid) -> bufA (1280,128,254), fused x2sum
    conv_gemm_wmma<128, 1, 1, 254, 254, 128, 1>
        <<<dim3(1280, 16), 32, 0, stream>>>(bufB, a2w, a2b, bufA, x2);

    p2_kernel<<<1, 256, 0, stream>>>(protos, p2b);

    // min_d -> out+128; P=200 split: tiles [0,7) and [7,13)
    proto_min_wmma<7, 0> <<<dim3(1280), 32, 0, stream>>>(bufA, protos, x2, p2b, out + 128);
    proto_min_wmma<6, 7> <<<dim3(1280), 32, 0, stream>>>(bufA, protos, x2, p2b, out + 128);

    // outputs -> out[0..127]
    final_kernel<<<dim3(64), 256, 0, stream>>>(out + 128, lastw, out);
}